// RIMAEStudent_38843684225805
// MI455X (gfx1250) — compile-verified
//
#include <hip/hip_runtime.h>

#define B_ 16
#define N_ 8192
#define G_ 512
#define M_ 32
#define CHUNK 2048
#define NCHUNKS (N_ / CHUNK)

typedef float v2f __attribute__((ext_vector_type(2)));
typedef float v8f __attribute__((ext_vector_type(8)));

// ---------------------------------------------------------------------------
// Kernel 1: deterministic farthest point sampling (start index 0).
// One block per batch. xyz cached SoA in LDS, running min-distance in LDS,
// two-level (wave shuffle + LDS) argmax per round. Writes center coords.
// ---------------------------------------------------------------------------
__global__ void fps_kernel(const float* __restrict__ xyz,
                           float* __restrict__ center)
{
    extern __shared__ unsigned char smraw[];
    float* px   = (float*)smraw;      // N_
    float* py   = px + N_;            // N_
    float* pz   = py + N_;            // N_
    float* dist = pz + N_;            // N_
    float* rv   = dist + N_;          // 32
    int*   ri   = (int*)(rv + 32);    // 32
    int*   bfar = ri + 32;            // 1

    const int b    = blockIdx.x;
    const int tid  = threadIdx.x;
    const int lane = tid & 31;
    const int wid  = tid >> 5;
    const float* xb = xyz + (size_t)b * N_ * 3;

    for (int i = tid; i < N_; i += blockDim.x) {
        px[i] = xb[i * 3 + 0];
        py[i] = xb[i * 3 + 1];
        pz[i] = xb[i * 3 + 2];
        dist[i] = 1e10f;
    }
    __syncthreads();

    int far = 0;
    for (int s = 0; s < G_; ++s) {
        float cx = px[far], cy = py[far], cz = pz[far];
        if (tid == 0) {
            float* co = center + ((size_t)b * G_ + s) * 3;
            co[0] = cx; co[1] = cy; co[2] = cz;
        }
        float mx = -1.0f; int mi = 0;
        for (int i = tid; i < N_; i += blockDim.x) {
            float dx = px[i] - cx, dy = py[i] - cy, dz = pz[i] - cz;
            float d  = dx * dx + dy * dy + dz * dz;
            float nd = fminf(dist[i], d);
            dist[i] = nd;
            if (nd > mx) { mx = nd; mi = i; }
        }
        // wave32 argmax (tie -> smaller index, matching jnp.argmax)
        for (int off = 16; off > 0; off >>= 1) {
            float ov = __shfl_xor(mx, off, 32);
            int   oi = __shfl_xor(mi, off, 32);
            if (ov > mx || (ov == mx && oi < mi)) { mx = ov; mi = oi; }
        }
        if (lane == 0) { rv[wid] = mx; ri[wid] = mi; }
        __syncthreads();
        if (wid == 0) {
            float v  = rv[lane];
            int   ix = ri[lane];
            for (int off = 16; off > 0; off >>= 1) {
                float ov = __shfl_xor(v, off, 32);
                int   oi = __shfl_xor(ix, off, 32);
                if (ov > v || (ov == v && oi < ix)) { v = ov; ix = oi; }
            }
            if (lane == 0) bfar[0] = ix;
        }
        __syncthreads();
        far = bfar[0];
    }
}

// ---------------------------------------------------------------------------
// Kernel 2: pairwise distances via V_WMMA_F32_16X16X4_F32 (K=3 padded to 4)
// + streaming per-row top-32 selection. One block per 16 centers.
// Stages gathered (xyz[knn] - center) into the canonical output region.
// ---------------------------------------------------------------------------
__global__ void knn_kernel(const float* __restrict__ xyz,
                           const float* __restrict__ center,
                           float* __restrict__ nbr_out)
{
    extern __shared__ unsigned char smraw[];
    float* sDist = (float*)smraw;          // 16 * CHUNK
    float* sCn   = sDist + 16 * CHUNK;     // 16 (|c|^2)
    float* sC    = sCn + 16;               // 16 * 3 (center coords)

    const int blk  = blockIdx.x;
    const int b    = blk >> 5;             // / (G_/16)
    const int gt   = blk & 31;
    const int g0   = gt * 16;
    const int tid  = threadIdx.x;
    const int lane = tid & 31;
    const int wid  = tid >> 5;
    const bool lo16 = (lane < 16);
    const float* xb = xyz + (size_t)b * N_ * 3;

    if (tid < 16) {
        const float* cp = center + ((size_t)b * G_ + g0 + tid) * 3;
        float x = cp[0], y = cp[1], z = cp[2];
        sC[tid * 3 + 0] = x; sC[tid * 3 + 1] = y; sC[tid * 3 + 2] = z;
        sCn[tid] = x * x + y * y + z * z;
    }
    __syncthreads();

    // A fragment: 16x4 f32. Lanes 0-15 hold (K0,K1)=(x,y) of row M=lane,
    // lanes 16-31 hold (K2,K3)=(z,0) of row M=lane-16. Branchless build.
    v2f afrag;
    {
        int m = lane & 15;
        float ax = sC[m * 3 + 0], ay = sC[m * 3 + 1], az = sC[m * 3 + 2];
        afrag.x = lo16 ? ax : az;
        afrag.y = lo16 ? ay : 0.0f;
    }
    // |c|^2 values this lane needs for D rows r+mhi (loop-invariant -> regs)
    const int mhi = (lane >> 4) << 3;
    float cn[8];
#pragma unroll
    for (int r = 0; r < 8; ++r) cn[r] = sCn[mhi + r];

    // per-lane carry: lane k holds the k-th best (val, idx) of rows wid, wid+8
    float carV0 = 1e30f, carV1 = 1e30f;
    int   carI0 = 0,     carI1 = 0;

    for (int ch = 0; ch < NCHUNKS; ++ch) {
        // ---- phase 1: distances for this chunk via WMMA (fully convergent) ----
        for (int t = wid; t < CHUNK / 16; t += 8) {
            int n0 = ch * CHUNK + t * 16;
            int p  = n0 + (lane & 15);
            // speculative prefetch of this wave's next tile (clamped, branchless)
            {
                int pnext = p + 16 * 8;
                pnext = (pnext < N_) ? pnext : (N_ - 1);
                __builtin_prefetch(xb + (size_t)pnext * 3, 0, 3);
            }
            // branchless fragment load: dword offset 0 -> (x,y), 1 -> (y,z)
            const float* pp = xb + (size_t)p * 3 + (lo16 ? 0 : 1);
            float q0 = pp[0], q1 = pp[1];
            float f0 = lo16 ? q0 : q1;          // x  or z
            float f1 = lo16 ? q1 : 0.0f;        // y  or 0
            v2f bfrag; bfrag.x = f0; bfrag.y = f1;
            float part = f0 * f0 + f1 * f1;
            float pn = part + __shfl_xor(part, 16, 32);  // |p|^2 for n=lane&15
            v8f acc = {0.f, 0.f, 0.f, 0.f, 0.f, 0.f, 0.f, 0.f};
            acc = __builtin_amdgcn_wmma_f32_16x16x4_f32(
                      false, afrag, false, bfrag, (short)0, acc, false, false);
            int nloc = t * 16 + (lane & 15);
#pragma unroll
            for (int r = 0; r < 8; ++r) {
                // D VGPR r: M = r (lanes 0-15) / r+8 (lanes 16-31)
                float dval = cn[r] + pn - 2.0f * acc[r];
                sDist[(mhi + r) * CHUNK + nloc] = dval;
            }
        }
        __syncthreads();

        // ---- phase 2: streaming top-32 per row (one wave per row) ----
#pragma unroll
        for (int rp = 0; rp < 2; ++rp) {
            int m = wid + rp * 8;
            volatile float* dv = sDist + m * CHUNK;
            float carV = rp ? carV1 : carV0;
            int   carI = rp ? carI1 : carI0;
            float nV = 1e30f; int nI = 0;
            for (int k = 0; k < M_; ++k) {
                float bv = 1e30f; int bp = CHUNK + 64;   // sentinel
                for (int pos = lane; pos < CHUNK; pos += 32) {
                    float v = dv[pos];
                    if (v < bv || (v == bv && pos < bp)) { bv = v; bp = pos; }
                }
                {   // this lane's carry candidate lives at position CHUNK+lane
                    int cpos = CHUNK + lane;
                    if (carV < bv || (carV == bv && cpos < bp)) { bv = carV; bp = cpos; }
                }
                for (int off = 16; off > 0; off >>= 1) {
                    float ov = __shfl_xor(bv, off, 32);
                    int   op = __shfl_xor(bp, off, 32);
                    if (ov < bv || (ov == bv && op < bp)) { bv = ov; bp = op; }
                }
                int srcLane = bp - CHUNK;
                int carWin  = __shfl(carI, (srcLane >= 0 && srcLane < 32) ? srcLane : 0, 32);
                int gidx    = (bp < CHUNK) ? (ch * CHUNK + bp) : carWin;
                if (lane == k) { nV = bv; nI = gidx; }
                // exclude the winner from the domain
                if (bp < CHUNK) {
                    if (lane == (bp & 31)) dv[bp] = 1e30f;
                } else {
                    if (lane == srcLane) carV = 1e30f;
                }
                __builtin_amdgcn_wave_barrier();
            }
            if (rp) { carV1 = nV; carI1 = nI; } else { carV0 = nV; carI0 = nI; }
        }
        __syncthreads();
    }

    // ---- output: gathered neighbor minus center (lane = neighbor slot k) ----
#pragma unroll
    for (int rp = 0; rp < 2; ++rp) {
        int m = wid + rp * 8;
        int g = g0 + m;
        int idx = rp ? carI1 : carI0;
        const float* pp = xb + (size_t)idx * 3;
        float nx = pp[0] - sC[m * 3 + 0];
        float ny = pp[1] - sC[m * 3 + 1];
        float nz = pp[2] - sC[m * 3 + 2];
        float* op = nbr_out + (((size_t)b * G_ + g) * M_ + lane) * 3;
        op[0] = nx; op[1] = ny; op[2] = nz;
    }
}

// ---------------------------------------------------------------------------
// Kernel 3: content/orientation disentanglement. One wave per group
// (lane = point). Gram via shuffle sums, redundant per-lane 3x3 Jacobi,
// canonical = flat @ V, sign fix from canonical mean.
// ---------------------------------------------------------------------------
__global__ void disent_kernel(float* __restrict__ canon,   // in: nbr, out: canonical
                              float* __restrict__ basisT)
{
    const int tid  = threadIdx.x;
    const int lane = tid & 31;
    const int wid  = tid >> 5;
    const int gid  = blockIdx.x * (blockDim.x >> 5) + wid;
    if (gid >= B_ * G_) return;

    float* base = canon + (size_t)gid * M_ * 3;
    float x = base[lane * 3 + 0];
    float y = base[lane * 3 + 1];
    float z = base[lane * 3 + 2];

    // mean over the 32 points
    float sx = x, sy = y, sz = z;
    for (int off = 16; off > 0; off >>= 1) {
        sx += __shfl_xor(sx, off, 32);
        sy += __shfl_xor(sy, off, 32);
        sz += __shfl_xor(sz, off, 32);
    }
    const float inv = 1.0f / (float)M_;
    float cx = x - sx * inv, cy = y - sy * inv, cz = z - sz * inv;

    // Gram of centered points
    float a00 = cx * cx, a01 = cx * cy, a02 = cx * cz;
    float a11 = cy * cy, a12 = cy * cz, a22 = cz * cz;
    for (int off = 16; off > 0; off >>= 1) {
        a00 += __shfl_xor(a00, off, 32);
        a01 += __shfl_xor(a01, off, 32);
        a02 += __shfl_xor(a02, off, 32);
        a11 += __shfl_xor(a11, off, 32);
        a12 += __shfl_xor(a12, off, 32);
        a22 += __shfl_xor(a22, off, 32);
    }

    // cyclic Jacobi eigendecomposition (redundant per lane)
    float A[3][3] = {{a00, a01, a02}, {a01, a11, a12}, {a02, a12, a22}};
    float V[3][3] = {{1.f, 0.f, 0.f}, {0.f, 1.f, 0.f}, {0.f, 0.f, 1.f}};
    for (int sweep = 0; sweep < 10; ++sweep) {
#pragma unroll
        for (int pair = 0; pair < 3; ++pair) {
            const int p = (pair == 2) ? 1 : 0;
            const int q = (pair == 0) ? 1 : 2;
            float apq = A[p][q];
            if (fabsf(apq) > 1e-18f) {
                float tau = (A[q][q] - A[p][p]) / (2.0f * apq);
                float t = ((tau >= 0.f) ? 1.f : -1.f) /
                          (fabsf(tau) + sqrtf(1.0f + tau * tau));
                float c = 1.0f / sqrtf(1.0f + t * t);
                float s = t * c;
                // A = J^T A J
                float b0p = c * A[0][p] - s * A[0][q], b0q = s * A[0][p] + c * A[0][q];
                float b1p = c * A[1][p] - s * A[1][q], b1q = s * A[1][p] + c * A[1][q];
                float b2p = c * A[2][p] - s * A[2][q], b2q = s * A[2][p] + c * A[2][q];
                A[0][p] = b0p; A[0][q] = b0q;
                A[1][p] = b1p; A[1][q] = b1q;
                A[2][p] = b2p; A[2][q] = b2q;
                float rp0 = c * A[p][0] - s * A[q][0], rq0 = s * A[p][0] + c * A[q][0];
                float rp1 = c * A[p][1] - s * A[q][1], rq1 = s * A[p][1] + c * A[q][1];
                float rp2 = c * A[p][2] - s * A[q][2], rq2 = s * A[p][2] + c * A[q][2];
                A[p][0] = rp0; A[p][1] = rp1; A[p][2] = rp2;
                A[q][0] = rq0; A[q][1] = rq1; A[q][2] = rq2;
                // V = V J
                float v0p = c * V[0][p] - s * V[0][q], v0q = s * V[0][p] + c * V[0][q];
                float v1p = c * V[1][p] - s * V[1][q], v1q = s * V[1][p] + c * V[1][q];
                float v2p = c * V[2][p] - s * V[2][q], v2q = s * V[2][p] + c * V[2][q];
                V[0][p] = v0p; V[0][q] = v0q;
                V[1][p] = v1p; V[1][q] = v1q;
                V[2][p] = v2p; V[2][q] = v2q;
            }
        }
    }

    // sort eigenpairs descending (columns of V)
    float w0 = A[0][0], w1 = A[1][1], w2 = A[2][2];
    float v00 = V[0][0], v10 = V[1][0], v20 = V[2][0];
    float v01 = V[0][1], v11 = V[1][1], v21 = V[2][1];
    float v02 = V[0][2], v12 = V[1][2], v22 = V[2][2];
    float tw, t0, t1, t2;
    if (w0 < w1) { tw = w0; w0 = w1; w1 = tw;
                   t0 = v00; v00 = v01; v01 = t0;
                   t1 = v10; v10 = v11; v11 = t1;
                   t2 = v20; v20 = v21; v21 = t2; }
    if (w0 < w2) { tw = w0; w0 = w2; w2 = tw;
                   t0 = v00; v00 = v02; v02 = t0;
                   t1 = v10; v10 = v12; v12 = t1;
                   t2 = v20; v20 = v22; v22 = t2; }
    if (w1 < w2) { tw = w1; w1 = w2; w2 = tw;
                   t0 = v01; v01 = v02; v02 = t0;
                   t1 = v11; v11 = v12; v12 = t1;
                   t2 = v21; v21 = v22; v22 = t2; }

    // canonical = flat @ basis (per lane)
    float c0 = x * v00 + y * v10 + z * v20;
    float c1 = x * v01 + y * v11 + z * v21;
    float c2 = x * v02 + y * v12 + z * v22;

    // sign from canonical mean (sign(0) -> +1)
    float m0 = c0, m1 = c1, m2 = c2;
    for (int off = 16; off > 0; off >>= 1) {
        m0 += __shfl_xor(m0, off, 32);
        m1 += __shfl_xor(m1, off, 32);
        m2 += __shfl_xor(m2, off, 32);
    }
    float s0 = (m0 < 0.f) ? -1.f : 1.f;
    float s1 = (m1 < 0.f) ? -1.f : 1.f;
    float s2 = (m2 < 0.f) ? -1.f : 1.f;

    base[lane * 3 + 0] = c0 * s0;
    base[lane * 3 + 1] = c1 * s1;
    base[lane * 3 + 2] = c2 * s2;

    // basisT = transpose(basis * s): basisT[i][j] = V[j][i] * s_i
    if (lane < 9) {
        const int i = lane / 3, j = lane % 3;
        float Vji;
        if (j == 0)      Vji = (i == 0) ? v00 : (i == 1) ? v01 : v02;
        else if (j == 1) Vji = (i == 0) ? v10 : (i == 1) ? v11 : v12;
        else             Vji = (i == 0) ? v20 : (i == 1) ? v21 : v22;
        float si = (i == 0) ? s0 : (i == 1) ? s1 : s2;
        basisT[(size_t)gid * 9 + lane] = Vji * si;
    }
}

// ---------------------------------------------------------------------------
extern "C" void kernel_launch(void* const* d_in, const int* in_sizes, int n_in,
                              void* d_out, int out_size, void* d_ws, size_t ws_size,
                              hipStream_t stream)
{
    (void)in_sizes; (void)n_in; (void)out_size; (void)d_ws; (void)ws_size;
    const float* xyz = (const float*)d_in[0];
    float* out    = (float*)d_out;
    float* canon  = out;                               // [16,512,32,3] = 786432
    float* basisT = out + (size_t)B_ * G_ * M_ * 3;    // [16,512,3,3]  =  73728
    float* center = basisT + (size_t)B_ * G_ * 9;      // [16,512,3]    =  24576

    size_t fps_lds = (size_t)(4 * N_) * sizeof(float) + 32 * sizeof(float)
                   + 33 * sizeof(int);
    fps_kernel<<<B_, 1024, fps_lds, stream>>>(xyz, center);

    size_t knn_lds = (size_t)(16 * CHUNK + 16 + 48) * sizeof(float);
    knn_kernel<<<B_ * (G_ / 16), 256, knn_lds, stream>>>(xyz, center, canon);

    disent_kernel<<<(B_ * G_) / 4, 128, 0, stream>>>(canon, basisT);
}